// TensorCP_63763084476735
// MI455X (gfx1250) — compile-verified
//
#include <hip/hip_runtime.h>

// ---------------------------------------------------------------------------
// TensorCP (TensoRF-style CP decomposition) forward for MI455X / gfx1250.
//
// out layout (float32, flat): [ cat_feature (N x 128) | uncertainty (N) ]
//   cat cols 0..63   = coarse feat @ Wc^T
//   cat cols 64..127 = fine   feat @ Wf^T
//
// One wave32 == one 16-point tile, GEMMs on V_WMMA_F32_16X16X4_F32 (exact
// fp32 matrix path; problem is output-bandwidth bound at ~23 us on 23.3 TB/s,
// so full-precision matrix ops are free).
//
// Round-1 restructure: K is the OUTER loop. Each lane computes only the two
// feature values its A-fragment needs at step k (ranks 4k+2h, 4k+2h+1 with
// h = lane>>4), then issues 4 WMMAs into 4 independent accumulator chains
// (dep distance 4, no fv[32] panel). This keeps VGPRs well under 256 (no
// s_set_vgpr_msb spam, better occupancy) and overlaps gathers with WMMAs.
// Weights sit in LDS with a 68-dword padded stride -> conflict-free
// ds_load_b64 across all 64 banks.
// ---------------------------------------------------------------------------

typedef __attribute__((ext_vector_type(2))) float v2f;
typedef __attribute__((ext_vector_type(8))) float v8f;

#define RANK 64
#define FDIM 64
#define DC   128
#define DF   512
#define WSTRIDE 68   // padded LDS row stride (dwords): lane n -> banks 4n+c{0..3}

__device__ __forceinline__ void mkidx(float c, int D, int& i0, int& i1, float& w) {
    float pos = c * (float)(D - 1);      // ((2c-1)+1)*0.5*(D-1) == c*(D-1)
    int i = (int)floorf(pos);
    i = max(0, min(i, D - 1));
    i0 = i;
    i1 = min(i + 1, D - 1);
    w = pos - (float)i;
}

__device__ __forceinline__ float lerp1(const float* __restrict__ base,
                                       int i0, int i1, float w) {
    float a = base[i0];
    float b = base[i1];
    return fmaf(w, b - a, a);
}

// 3-dim CP product for one rank of one table (table layout (3, RANK, D)).
__device__ __forceinline__ float cp3(const float* __restrict__ t, int r, int D,
                                     int i0x, int i1x, float wx,
                                     int i0y, int i1y, float wy,
                                     int i0z, int i1z, float wz) {
    const float* bx = t + (0 * RANK + r) * D;
    const float* by = t + (1 * RANK + r) * D;
    const float* bz = t + (2 * RANK + r) * D;
    return lerp1(bx, i0x, i1x, wx)
         * lerp1(by, i0y, i1y, wy)
         * lerp1(bz, i0z, i1z, wz);
}

__global__ __launch_bounds__(128) void tensorcp_wmma_kernel(
    const float* __restrict__ xyz,     // (N,3) in [0,1]
    const float* __restrict__ line_c,  // (3,RANK,DC)
    const float* __restrict__ line_f,  // (3,RANK,DF)
    const float* __restrict__ ulines,  // (3,RANK,DC)
    const float* __restrict__ Wc,      // (FDIM,RANK) row-major
    const float* __restrict__ Wf,      // (FDIM,RANK) row-major
    float* __restrict__ out,           // N*128 + N floats
    int N, int ntiles)
{
    __shared__ float sWc[FDIM * WSTRIDE];
    __shared__ float sWf[FDIM * WSTRIDE];

    const int tid = threadIdx.x;

    // Stage both 64x64 weight matrices into LDS (padded rows).
    for (int idx = tid; idx < FDIM * RANK; idx += 128) {
        int row = idx >> 6;
        int col = idx & 63;
        sWc[row * WSTRIDE + col] = Wc[idx];
        sWf[row * WSTRIDE + col] = Wf[idx];
    }
    __syncthreads();

    const int lane = tid & 31;
    const int wave = tid >> 5;
    int tile = blockIdx.x * 4 + wave;
    const bool active = (tile < ntiles);
    if (!active) tile = ntiles - 1;          // keep EXEC full for WMMA
    const int ptbase = tile * 16;

    const int p = lane & 15;                 // point row within tile (== N col of B/D)
    const int h = lane >> 4;                 // K-pair half selector
    const int pg = min(ptbase + p, N - 1);

    const float x = xyz[pg * 3 + 0];
    const float y = xyz[pg * 3 + 1];
    const float z = xyz[pg * 3 + 2];

    int i0cx, i1cx, i0cy, i1cy, i0cz, i1cz;
    int i0fx, i1fx, i0fy, i1fy, i0fz, i1fz;
    float wcx, wcy, wcz, wfx, wfy, wfz;
    mkidx(x, DC, i0cx, i1cx, wcx);
    mkidx(y, DC, i0cy, i1cy, wcy);
    mkidx(z, DC, i0cz, i1cz, wcz);
    mkidx(x, DF, i0fx, i1fx, wfx);
    mkidx(y, DF, i0fy, i1fy, wfy);
    mkidx(z, DF, i0fz, i1fz, wfz);

    // =============== coarse GEMM (K outer, 4 independent acc chains) =======
    float usum = 0.0f;
    {
        v8f acc[4] = {};
        #pragma unroll
        for (int k = 0; k < 16; ++k) {
            const int r0 = 4 * k + 2 * h;    // the two ranks this lane's
            const int r1 = r0 + 1;           // A-fragment carries at step k
            v2f a;
            a.x = cp3(line_c, r0, DC, i0cx, i1cx, wcx, i0cy, i1cy, wcy, i0cz, i1cz, wcz);
            a.y = cp3(line_c, r1, DC, i0cx, i1cx, wcx, i0cy, i1cy, wcy, i0cz, i1cz, wcz);
            // uncertainty shares the coarse interpolation indices
            usum += cp3(ulines, r0, DC, i0cx, i1cx, wcx, i0cy, i1cy, wcy, i0cz, i1cz, wcz);
            usum += cp3(ulines, r1, DC, i0cx, i1cx, wcx, i0cy, i1cy, wcy, i0cz, i1cz, wcz);
            #pragma unroll
            for (int fb = 0; fb < 4; ++fb) {
                const float* wp = &sWc[(fb * 16 + p) * WSTRIDE + 4 * k + 2 * h];
                v2f b = *(const v2f*)wp;
                acc[fb] = __builtin_amdgcn_wmma_f32_16x16x4_f32(
                    false, a, false, b, (short)0, acc[fb], false, false);
            }
        }
        #pragma unroll
        for (int fb = 0; fb < 4; ++fb) {
            #pragma unroll
            for (int j = 0; j < 8; ++j) {
                const int row = ptbase + j + 8 * h;  // D: VGPR j -> M = j + 8*half
                if (active && row < N)
                    out[(long long)row * 128 + fb * 16 + p] = acc[fb][j];
            }
        }
    }

    // =============== fine GEMM =============================================
    {
        v8f acc[4] = {};
        #pragma unroll
        for (int k = 0; k < 16; ++k) {
            const int r0 = 4 * k + 2 * h;
            const int r1 = r0 + 1;
            v2f a;
            a.x = cp3(line_f, r0, DF, i0fx, i1fx, wfx, i0fy, i1fy, wfy, i0fz, i1fz, wfz);
            a.y = cp3(line_f, r1, DF, i0fx, i1fx, wfx, i0fy, i1fy, wfy, i0fz, i1fz, wfz);
            #pragma unroll
            for (int fb = 0; fb < 4; ++fb) {
                const float* wp = &sWf[(fb * 16 + p) * WSTRIDE + 4 * k + 2 * h];
                v2f b = *(const v2f*)wp;
                acc[fb] = __builtin_amdgcn_wmma_f32_16x16x4_f32(
                    false, a, false, b, (short)0, acc[fb], false, false);
            }
        }
        #pragma unroll
        for (int fb = 0; fb < 4; ++fb) {
            #pragma unroll
            for (int j = 0; j < 8; ++j) {
                const int row = ptbase + j + 8 * h;
                if (active && row < N)
                    out[(long long)row * 128 + 64 + fb * 16 + p] = acc[fb][j];
            }
        }
    }

    // =============== uncertainty: pairwise rank-half reduce =================
    const float utot = usum + __shfl_xor(usum, 16);
    if (active && lane < 16) {
        const int row = ptbase + p;
        if (row < N)
            out[(long long)N * 128 + row] = utot;
    }
}

extern "C" void kernel_launch(void* const* d_in, const int* in_sizes, int n_in,
                              void* d_out, int out_size, void* d_ws, size_t ws_size,
                              hipStream_t stream) {
    const float* xyz = (const float*)d_in[0];
    const float* lc  = (const float*)d_in[1];
    const float* lf  = (const float*)d_in[2];
    const float* ul  = (const float*)d_in[3];
    const float* Wc  = (const float*)d_in[4];
    const float* Wf  = (const float*)d_in[5];
    float* out = (float*)d_out;

    const int N = in_sizes[0] / 3;             // (N,3) coords
    const int ntiles = (N + 15) / 16;          // 16 points per wave
    const int blocks = (ntiles + 3) / 4;       // 4 waves (128 threads) per block

    tensorcp_wmma_kernel<<<blocks, 128, 0, stream>>>(xyz, lc, lf, ul, Wc, Wf,
                                                     out, N, ntiles);
}